// RL_GCN_37744172597441
// MI455X (gfx1250) — compile-verified
//
#include <hip/hip_runtime.h>

typedef __attribute__((ext_vector_type(2))) float v2f;
typedef __attribute__((ext_vector_type(8))) float v8f;

#define F_DIM 128
#define C_DIM 64

__device__ __forceinline__ v8f wmma4(v2f a, v2f b, v8f c) {
  // D = A(16x4 f32) * B(4x16 f32) + C(16x16 f32)  -> v_wmma_f32_16x16x4_f32
  return __builtin_amdgcn_wmma_f32_16x16x4_f32(false, a, false, b, (short)0, c,
                                               false, false);
}

__global__ void zero_f32_kernel(float* __restrict__ p, long long n) {
  long long i = (long long)blockIdx.x * blockDim.x + threadIdx.x;
  if (i < n) p[i] = 0.0f;
}

__global__ void degree_kernel(const int* __restrict__ dst, float* __restrict__ deg, int E) {
  int e = blockIdx.x * blockDim.x + threadIdx.x;
  if (e < E) unsafeAtomicAdd(&deg[dst[e]], 1.0f);
}

// One 32-lane group per edge; each lane adds 4 contiguous floats (native f32 L2 atomics).
__global__ void scatter_sum_kernel(const float* __restrict__ feat,
                                   const int* __restrict__ src,
                                   const int* __restrict__ dst,
                                   float* __restrict__ agg, int E) {
  long long t = (long long)blockIdx.x * blockDim.x + threadIdx.x;
  int e = (int)(t >> 5);
  int g = (int)(t & 31);
  if (e >= E) return;
  int s = src[e];
  int d = dst[e];
  const float4 v = *reinterpret_cast<const float4*>(feat + (size_t)s * F_DIM + g * 4);
  float* o = agg + (size_t)d * F_DIM + g * 4;
  unsafeAtomicAdd(o + 0, v.x);
  unsafeAtomicAdd(o + 1, v.y);
  unsafeAtomicAdd(o + 2, v.z);
  unsafeAtomicAdd(o + 3, v.w);
}

// Pair-interleaved LDS index for weight W[k][c] with NCOL columns:
// s[(k>>1)*2*NCOL + c*2 + (k&1)]  ->  {W[kk][c], W[kk+1][c]} is one aligned v2f.
__device__ __forceinline__ int swz(int k, int c, int ncol) {
  return ((k >> 1) * ncol + c) * 2 + (k & 1);
}

// out[tile rows] = act( (agg/max(deg,1)) @ wl + x @ wr + bias )
// One wave per 16-node tile, full 128-wide output (8 column tiles of 16).
template <bool RELU>
__global__ void sage_layer_kernel(const float* __restrict__ agg,
                                  const float* __restrict__ deg,
                                  const float* __restrict__ x,
                                  const float* __restrict__ wl,
                                  const float* __restrict__ wr,
                                  const float* __restrict__ bias,
                                  float* __restrict__ out,
                                  int num_tiles) {
  extern __shared__ float smem[];
  float* s_wl = smem;                   // 128x128 f32 = 64 KB, pair-interleaved
  float* s_wr = smem + F_DIM * F_DIM;   // 128x128 f32 = 64 KB, pair-interleaved

  for (int idx = threadIdx.x; idx < F_DIM * F_DIM; idx += blockDim.x) {
    const int k = idx >> 7;         // row (K)
    const int c = idx & (F_DIM - 1);
    const int si = swz(k, c, F_DIM);
    s_wl[si] = wl[idx];
    s_wr[si] = wr[idx];
  }
  __syncthreads();

  const int tile = blockIdx.x * (blockDim.x >> 5) + (threadIdx.x >> 5);
  if (tile >= num_tiles) return;  // wave-uniform: EXEC stays all-ones for WMMA

  const int lane  = threadIdx.x & 31;
  const int col   = lane & 15;    // A row within tile / B,C column
  const int khalf = lane >> 4;    // K sub-offset: lanes 0-15 -> K{0,1}, 16-31 -> K{2,3}
  const int row0  = tile * 16;
  const int row   = row0 + col;

  const float invd = 1.0f / fmaxf(deg[row], 1.0f);
  const float* aggRow = agg + (size_t)row * F_DIM;
  const float* xRow   = x   + (size_t)row * F_DIM;

  v8f acc[8];
#pragma unroll
  for (int nt = 0; nt < 8; ++nt) {
    const float bv = bias[nt * 16 + col];
#pragma unroll
    for (int r = 0; r < 8; ++r) acc[nt][r] = bv;
  }

  for (int k0 = 0; k0 < F_DIM; k0 += 4) {
    const int kk = k0 + 2 * khalf;
    v2f a1 = *reinterpret_cast<const v2f*>(aggRow + kk);
    a1 = a1 * invd;  // fuse mean into A fragment
    const v2f a2 = *reinterpret_cast<const v2f*>(xRow + kk);
    // pair-row base for this lane's K half; per-nt offsets are immediates
    const int pbase = ((kk >> 1) * F_DIM + col) * 2;
    const float* pl = s_wl + pbase;
    const float* pr = s_wr + pbase;
#pragma unroll
    for (int nt = 0; nt < 8; ++nt) {
      const v2f b1 = *reinterpret_cast<const v2f*>(pl + nt * 32);  // ds_load_b64
      const v2f b2 = *reinterpret_cast<const v2f*>(pr + nt * 32);  // ds_load_b64
      acc[nt] = wmma4(a1, b1, acc[nt]);
      acc[nt] = wmma4(a2, b2, acc[nt]);
    }
  }

  const int rbase = row0 + 8 * khalf;  // C/D: vgpr r -> row r (lanes 0-15) / r+8 (16-31)
#pragma unroll
  for (int nt = 0; nt < 8; ++nt) {
#pragma unroll
    for (int r = 0; r < 8; ++r) {
      float f = acc[nt][r];
      if (RELU) f = fmaxf(f, 0.0f);
      out[(size_t)(rbase + r) * F_DIM + nt * 16 + col] = f;
    }
  }
}

__global__ void final_linear_kernel(const float* __restrict__ h,
                                    const float* __restrict__ w,
                                    const float* __restrict__ bias,
                                    float* __restrict__ out,
                                    int num_tiles) {
  extern __shared__ float smem[];  // 128x64 f32 = 32 KB, pair-interleaved
  for (int idx = threadIdx.x; idx < F_DIM * C_DIM; idx += blockDim.x) {
    const int k = idx >> 6;          // row (K), 64 cols
    const int c = idx & (C_DIM - 1);
    smem[swz(k, c, C_DIM)] = w[idx];
  }
  __syncthreads();

  const int tile = blockIdx.x * (blockDim.x >> 5) + (threadIdx.x >> 5);
  if (tile >= num_tiles) return;

  const int lane  = threadIdx.x & 31;
  const int col   = lane & 15;
  const int khalf = lane >> 4;
  const int row0  = tile * 16;
  const float* hRow = h + (size_t)(row0 + col) * F_DIM;

  v8f acc[4];
#pragma unroll
  for (int nt = 0; nt < 4; ++nt) {
    const float bv = bias[nt * 16 + col];
#pragma unroll
    for (int r = 0; r < 8; ++r) acc[nt][r] = bv;
  }

  for (int k0 = 0; k0 < F_DIM; k0 += 4) {
    const int kk = k0 + 2 * khalf;
    const v2f a = *reinterpret_cast<const v2f*>(hRow + kk);
    const float* pw = smem + ((kk >> 1) * C_DIM + col) * 2;
#pragma unroll
    for (int nt = 0; nt < 4; ++nt) {
      const v2f b = *reinterpret_cast<const v2f*>(pw + nt * 32);  // ds_load_b64
      acc[nt] = wmma4(a, b, acc[nt]);
    }
  }

  const int rbase = row0 + 8 * khalf;
#pragma unroll
  for (int nt = 0; nt < 4; ++nt)
#pragma unroll
    for (int r = 0; r < 8; ++r)
      out[(size_t)(rbase + r) * C_DIM + nt * 16 + col] = acc[nt][r];
}

extern "C" void kernel_launch(void* const* d_in, const int* in_sizes, int n_in,
                              void* d_out, int out_size, void* d_ws, size_t ws_size,
                              hipStream_t stream) {
  const float* x     = (const float*)d_in[0];
  const int*   edge  = (const int*)d_in[1];   // [2,E]: row0 = src, row1 = dst (int32 after JAX canonicalization)
  const float* w1_l  = (const float*)d_in[2];
  const float* w1_r  = (const float*)d_in[3];
  const float* b1    = (const float*)d_in[4];
  const float* w2_l  = (const float*)d_in[5];
  const float* w2_r  = (const float*)d_in[6];
  const float* b2    = (const float*)d_in[7];
  const float* w_lin = (const float*)d_in[8];
  const float* b_lin = (const float*)d_in[9];
  float* out = (float*)d_out;

  const int N = in_sizes[0] / F_DIM;
  const int E = in_sizes[1] / 2;
  const int* src = edge;
  const int* dst = edge + E;

  // Workspace: deg[N] | bufA[N*128] (agg) | bufB[N*128] (hidden)  ~103 MB
  float* deg  = (float*)d_ws;
  float* bufA = deg + N;
  float* bufB = bufA + (size_t)N * F_DIM;

  const int num_tiles = N / 16;              // N=100000 -> 6250 exact
  const int WAVES_PER_BLOCK = 8;             // 256 threads, wave32
  dim3 blk(256);
  dim3 gemm_grid((num_tiles + WAVES_PER_BLOCK - 1) / WAVES_PER_BLOCK);
  const size_t lds_layer = 2u * F_DIM * F_DIM * sizeof(float);  // 128 KB (<320 KB WGP LDS)
  const size_t lds_final = (size_t)F_DIM * C_DIM * sizeof(float);

  // 1) zero deg + agg (contiguous), degree, scatter x -> agg
  long long zn = (long long)N * (F_DIM + 1);
  zero_f32_kernel<<<(int)((zn + 255) / 256), 256, 0, stream>>>(deg, zn);
  degree_kernel<<<(E + 255) / 256, 256, 0, stream>>>(dst, deg, E);
  long long st = (long long)E * 32;
  scatter_sum_kernel<<<(int)((st + 255) / 256), 256, 0, stream>>>(x, src, dst, bufA, E);

  // 2) layer 1: bufB = relu(mean(agg)@w1_l + x@w1_r + b1)
  sage_layer_kernel<true><<<gemm_grid, blk, lds_layer, stream>>>(
      bufA, deg, x, w1_l, w1_r, b1, bufB, num_tiles);

  // 3) re-aggregate h1, layer 2 in place (row tiles are independent)
  long long za = (long long)N * F_DIM;
  zero_f32_kernel<<<(int)((za + 255) / 256), 256, 0, stream>>>(bufA, za);
  scatter_sum_kernel<<<(int)((st + 255) / 256), 256, 0, stream>>>(bufB, src, dst, bufA, E);
  sage_layer_kernel<true><<<gemm_grid, blk, lds_layer, stream>>>(
      bufA, deg, bufB, w2_l, w2_r, b2, bufB, num_tiles);

  // 4) head: out = h2 @ w_lin + b_lin
  final_linear_kernel<<<gemm_grid, blk, lds_final, stream>>>(
      bufB, w_lin, b_lin, out, num_tiles);

  (void)n_in; (void)out_size; (void)ws_size;
}